// ProtoMIL_84997402788393
// MI455X (gfx1250) — compile-verified
//
#include <hip/hip_runtime.h>
#include <math.h>

// ---------------------------------------------------------------------------
// ProtoMIL forward for MI455X (gfx1250, wave32).
// Phase 1 (memory-bound, 512MB stream): scalar-score mat-vec at HBM rate.
// Phase 2 (embedding GEMM): V_WMMA_F32_16X16X4_F32 tiles.
// Phase 3: tiny epilogue in one workgroup.
// ---------------------------------------------------------------------------

typedef float v2f __attribute__((ext_vector_type(2)));
typedef float v8f __attribute__((ext_vector_type(8)));

#define N_ROWS  65536
#define F_DIM   2048
#define H_DIM   256
#define C_PROTO 64
#define K_HID   16
#define TOPK_N  10
#define EPS_F   1e-6f

// workspace layout (float units)
#define OFF_W       0
#define OFF_SCORES  2048
#define OFF_TOPIDX  (2048 + 65536)
#define OFF_E       (OFF_TOPIDX + 16)   // 80x256 padded embedding matrix
#define E_ROWS      80

// ---- 1) w[f] = W3[f,1] - W3[f,0]  (softmax(2) top-k == top-k of logit diff)
__global__ void prep_w_kernel(const float* __restrict__ W3, float* __restrict__ w) {
    int t = blockIdx.x * blockDim.x + threadIdx.x;
    if (t < F_DIM) w[t] = W3[2 * t + 1] - W3[2 * t];
}

// ---- 2) scores[n] = dot(x_path[n,:], w)  — one row per wave, float4 stream
__global__ void score_kernel(const float* __restrict__ x,
                             const float* __restrict__ w,
                             float* __restrict__ scores) {
    const int wave = threadIdx.x >> 5;
    const int lane = threadIdx.x & 31;
    const int row  = blockIdx.x * 8 + wave;
    const float4* xr = reinterpret_cast<const float4*>(x + (size_t)row * F_DIM);
    const float4* wv = reinterpret_cast<const float4*>(w);
    float acc = 0.f;
#pragma unroll 4
    for (int i = 0; i < F_DIM / 128; ++i) {
        float4 a = xr[lane + 32 * i];
        float4 b = wv[lane + 32 * i];
        acc = fmaf(a.x, b.x, fmaf(a.y, b.y, fmaf(a.z, b.z, fmaf(a.w, b.w, acc))));
    }
#pragma unroll
    for (int off = 16; off > 0; off >>= 1) acc += __shfl_xor(acc, off, 32);
    if (lane == 0) scores[row] = acc;
}

// ---- 3) iterative top-10 argmax, single 1024-thread block
__global__ void topk_kernel(float* __restrict__ scores, int* __restrict__ topidx) {
    __shared__ float sval[1024];
    __shared__ int   sidx[1024];
    const int tid = threadIdx.x;
    const float NEG = -__builtin_inff();
    for (int k = 0; k < TOPK_N; ++k) {
        float bv = NEG; int bi = 0;
        for (int idx = tid; idx < N_ROWS; idx += 1024) {
            float v = scores[idx];
            if (v > bv) { bv = v; bi = idx; }       // strict > : lowest index wins ties
        }
        sval[tid] = bv; sidx[tid] = bi;
        __syncthreads();
        for (int s = 512; s > 0; s >>= 1) {
            if (tid < s) {
                float ov = sval[tid + s]; int oi = sidx[tid + s];
                if (ov > sval[tid] || (ov == sval[tid] && oi < sidx[tid])) {
                    sval[tid] = ov; sidx[tid] = oi;
                }
            }
            __syncthreads();
        }
        if (tid == 0) { topidx[k] = sidx[0]; scores[sidx[0]] = NEG; }
        __syncthreads();
    }
}

// ---- 4) E = [x_path[top_idx]; prototype] @ W2 + b2   via WMMA f32 16x16x4
// One wave per (M-tile, N-tile). A layout (ISA 7.12.2, 32-bit 16x4):
//   VGPR0: K=0 (lanes 0-15) / K=2 (lanes 16-31); VGPR1: K=1 / K=3.
// B (4x16) mirrors: VGPR0 row K=0/K=2 striped over N=lane; VGPR1 K=1/K=3.
// D (16x16 f32): VGPR j -> M=j (lanes 0-15), M=j+8 (lanes 16-31), N=lane%16.
__global__ void gemm_wmma_kernel(const float* __restrict__ x,
                                 const float* __restrict__ proto,
                                 const float* __restrict__ W2,
                                 const float* __restrict__ b2,
                                 const int* __restrict__ topidx,
                                 float* __restrict__ E) {
    const int lane  = threadIdx.x;       // 0..31, full wave -> EXEC all ones
    const int laneM = lane & 15;
    const int hi    = lane >> 4;         // 0 or 1 (K half)
    const int nt    = blockIdx.x;        // 16 N-tiles
    const int mt    = blockIdx.y;        // 5 M-tiles (80 padded rows)
    const int r     = mt * 16 + laneM;

    const float* rowp; float ascale;
    if (r < TOPK_N)                { rowp = x     + (size_t)topidx[r] * F_DIM; ascale = 1.f; }
    else if (r < TOPK_N + C_PROTO) { rowp = proto + (size_t)(r - TOPK_N) * F_DIM; ascale = 1.f; }
    else                           { rowp = proto; ascale = 0.f; }   // zero-pad rows, no divergence

    const int n = nt * 16 + laneM;
    const float* bp = W2 + n;            // column n of W2 (row-major [2048,256])
    v8f c = {};
    for (int kk = 0; kk < F_DIM / 4; ++kk) {
        const int k0 = kk * 4 + hi * 2;
        float2 av = *reinterpret_cast<const float2*>(rowp + k0);
        v2f a; a.x = av.x * ascale; a.y = av.y * ascale;
        v2f b; b.x = bp[(size_t)k0 * H_DIM]; b.y = bp[(size_t)(k0 + 1) * H_DIM];
        c = __builtin_amdgcn_wmma_f32_16x16x4_f32(false, a, false, b,
                                                  (short)0, c, false, false);
    }
    const float bias = b2[n];
#pragma unroll
    for (int j = 0; j < 8; ++j) {
        const int row = mt * 16 + hi * 8 + j;
        E[(size_t)row * H_DIM + n] = c[j] + bias;
    }
}

// ---- 5) distances -> normalize -> mean -> rho -> relu -> classifier -> out
__global__ void finish_kernel(const float* __restrict__ E,
                              const float* __restrict__ Wrho,
                              const float* __restrict__ brho,
                              const float* __restrict__ Wc,
                              const float* __restrict__ bc,
                              float* __restrict__ out) {
    __shared__ float sim[TOPK_N][C_PROTO];
    __shared__ float rowmax[TOPK_N];
    __shared__ float sc[C_PROTO];
    __shared__ float hid[K_HID];
    __shared__ float bl[2];
    const int tid = threadIdx.x;

    for (int pair = tid; pair < TOPK_N * C_PROTO; pair += 256) {
        const int i = pair / C_PROTO, j = pair % C_PROTO;
        const float* fr = E + (size_t)i * H_DIM;
        const float* pr = E + (size_t)(TOPK_N + j) * H_DIM;
        float s = 0.f;
        for (int h = 0; h < H_DIM; ++h) {
            float d = fr[h] - pr[h] + EPS_F;
            s = fmaf(d, d, s);
        }
        sim[i][j] = sqrtf(s);
    }
    __syncthreads();
    if (tid < TOPK_N) {
        float m = sim[tid][0];
        for (int j = 1; j < C_PROTO; ++j) m = fmaxf(m, sim[tid][j]);
        rowmax[tid] = m;
    }
    __syncthreads();
    if (tid < C_PROTO) {
        float s = 0.f;
        for (int i = 0; i < TOPK_N; ++i) s += sim[i][tid] / rowmax[i];
        sc[tid] = s * (1.f / TOPK_N);
        out[5 + tid] = sc[tid];                       // sim_coding [1,64]
    }
    __syncthreads();
    if (tid < K_HID) {
        float h = brho[tid];
        for (int cI = 0; cI < C_PROTO; ++cI) h = fmaf(sc[cI], Wrho[cI * K_HID + tid], h);
        hid[tid] = fmaxf(h, 0.f);                     // relu
    }
    __syncthreads();
    if (tid < 2) {
        float v = bc[tid];
        for (int k = 0; k < K_HID; ++k) v = fmaf(hid[k], Wc[k * 2 + tid], v);
        bl[tid] = v;
    }
    __syncthreads();
    if (tid == 0) {
        float m = fmaxf(bl[0], bl[1]);
        float e0 = __expf(bl[0] - m), e1 = __expf(bl[1] - m);
        float z = e0 + e1;
        out[0] = bl[0]; out[1] = bl[1];               // bag_logits
        out[2] = e0 / z; out[3] = e1 / z;             // Y_prob
        out[4] = (e1 / z > e0 / z) ? 1.f : 0.f;       // Y_hat (argmax, ties->0)
    }
}

extern "C" void kernel_launch(void* const* d_in, const int* in_sizes, int n_in,
                              void* d_out, int out_size, void* d_ws, size_t ws_size,
                              hipStream_t stream) {
    (void)in_sizes; (void)n_in; (void)out_size; (void)ws_size;
    const float* x_path = (const float*)d_in[0];
    const float* proto  = (const float*)d_in[1];
    const float* W3     = (const float*)d_in[2];
    // d_in[3] = b3: constant shift, does not change top-k ordering -> unused
    const float* W2     = (const float*)d_in[4];
    const float* b2     = (const float*)d_in[5];
    const float* Wrho   = (const float*)d_in[6];
    const float* brho   = (const float*)d_in[7];
    const float* Wc     = (const float*)d_in[8];
    const float* bc     = (const float*)d_in[9];

    float* ws      = (float*)d_ws;
    float* w       = ws + OFF_W;
    float* scores  = ws + OFF_SCORES;
    int*   topidx  = (int*)(ws + OFF_TOPIDX);
    float* E       = ws + OFF_E;
    float* out     = (float*)d_out;

    prep_w_kernel<<<(F_DIM + 255) / 256, 256, 0, stream>>>(W3, w);
    score_kernel<<<N_ROWS / 8, 256, 0, stream>>>(x_path, w, scores);
    topk_kernel<<<1, 1024, 0, stream>>>(scores, topidx);
    dim3 gemm_grid(H_DIM / 16, E_ROWS / 16);          // (16, 5)
    gemm_wmma_kernel<<<gemm_grid, 32, 0, stream>>>(x_path, proto, W2, b2, topidx, E);
    finish_kernel<<<1, 256, 0, stream>>>(E, Wrho, brho, Wc, bc, out);
}